// MoE_16381005266955
// MI455X (gfx1250) — compile-verified
//
#include <hip/hip_runtime.h>
#include <hip/hip_bf16.h>

typedef __attribute__((ext_vector_type(2))) float v2f;
typedef __attribute__((ext_vector_type(8))) float v8f;

#define N_TOK   32768
#define D_IN    256
#define N_EXP   32
#define D_OUT   256
#define NOISE_S 0.1f
#define CHUNKS  8          // k_expert grid.x: tile chunks per expert

// ---------------- workspace layout (bytes) ----------------
// [0)        topk_idx   int[2*N]
// [256K)     topk_prob  f32[2*N]
// [512K)     assign_tok int[2*N]   (slot0 bucket region [0,N), slot1 [N,2N))
// [768K)     assign_prob f32[2*N]
// [1M)       counts  int[64]   (slot*32+e)
// [1M+256)   offsets int[64]
// [1M+512)   cursors int[64]
// [1M+768)   probsum_part f32[256*32]
// ----------------------------------------------------------

__global__ __launch_bounds__(64) void k_init(int* counts) {
    if (threadIdx.x < 64) counts[threadIdx.x] = 0;
}

// ---------------- gating: WMMA logits + softmax + top2 ----------------
__global__ __launch_bounds__(256) void k_gate(const float* __restrict__ x,
                                              const float* __restrict__ noise,
                                              const float* __restrict__ gw,
                                              const float* __restrict__ gb,
                                              int* __restrict__ topk_idx,
                                              float* __restrict__ topk_prob,
                                              int* __restrict__ counts,
                                              float* __restrict__ probsum_part) {
    __shared__ float lg[8][16][33];   // padded: stride 33 -> conflict-free column reads
    __shared__ float ps[32];
    const int tid = threadIdx.x;
    const int w   = tid >> 5;
    const int l   = tid & 31;
    if (tid < 32) ps[tid] = 0.0f;
    __syncthreads();

    const int g    = blockIdx.x * 8 + w;   // 16-token group id
    const int t0   = g * 16;
    const int half = l >> 4;
    const int col  = l & 15;

    // A: 16x4 fp32 tile of x; B: 4x16 slices of gate_w^T (two N-tiles -> 32 experts)
    const float* pX  = x  + (size_t)(t0 + col) * D_IN + 2 * half;
    const float* pB0 = gw + (size_t)col * D_IN + 2 * half;
    const float* pB1 = pB0 + 16 * D_IN;

    v8f c0 = {}; v8f c1 = {};
    for (int kk = 0; kk < D_IN / 4; ++kk) {
        v2f a  = *(const v2f*)(pX  + kk * 4);
        v2f b0 = *(const v2f*)(pB0 + kk * 4);
        v2f b1 = *(const v2f*)(pB1 + kk * 4);
        c0 = __builtin_amdgcn_wmma_f32_16x16x4_f32(false, a, false, b0, (short)0, c0, false, false);
        c1 = __builtin_amdgcn_wmma_f32_16x16x4_f32(false, a, false, b1, (short)0, c1, false, false);
    }
#pragma unroll
    for (int r = 0; r < 8; ++r) {
        const int M = r + 8 * half;           // token row within group
        lg[w][M][col]      = c0[r];           // experts 0..15
        lg[w][M][col + 16] = c1[r];           // experts 16..31
    }
    __syncthreads();

    if (l < 16) {                              // lane l owns token t0+l
        const int t = t0 + l;
        const float* nz = noise + (size_t)t * N_EXP;
        float lv[N_EXP];
        float mx = -3.0e38f;
#pragma unroll
        for (int e = 0; e < N_EXP; ++e) {
            float v = lg[w][l][e] + gb[e] + NOISE_S * nz[e];
            lv[e] = v;
            mx = fmaxf(mx, v);
        }
        float sum = 0.0f;
#pragma unroll
        for (int e = 0; e < N_EXP; ++e) { float ev = expf(lv[e] - mx); lv[e] = ev; sum += ev; }
        const float inv = 1.0f / sum;
        // top-2 (lowest index wins ties, matching lax.top_k)
        int i0 = 0; float v0 = lv[0];
#pragma unroll
        for (int e = 1; e < N_EXP; ++e) if (lv[e] > v0) { v0 = lv[e]; i0 = e; }
        int i1 = (i0 == 0) ? 1 : 0; float v1 = lv[i1];
#pragma unroll
        for (int e = 0; e < N_EXP; ++e) if (e != i0 && lv[e] > v1) { v1 = lv[e]; i1 = e; }

        topk_idx [2 * t]     = i0;
        topk_idx [2 * t + 1] = i1;
        topk_prob[2 * t]     = v0 * inv;
        topk_prob[2 * t + 1] = v1 * inv;
        atomicAdd(&counts[i0], 1);
        atomicAdd(&counts[32 + i1], 1);
#pragma unroll
        for (int e = 0; e < N_EXP; ++e) atomicAdd(&ps[e], lv[e] * inv);  // ds_add_f32
    }
    __syncthreads();
    if (tid < 32) probsum_part[(size_t)blockIdx.x * 32 + tid] = ps[tid];  // deterministic partials
}

// ---------------- prefix-scan of bucket counts + load-balance loss ----------------
__global__ __launch_bounds__(64) void k_scan(const int* __restrict__ counts,
                                             int* __restrict__ offsets,
                                             int* __restrict__ cursors,
                                             const float* __restrict__ probsum_part,
                                             float* __restrict__ out_loss) {
    __shared__ float m[32];
    const int tid = threadIdx.x;
    if (tid < 32) {
        float s = 0.0f;
#pragma unroll 8
        for (int b = 0; b < 256; ++b) s += probsum_part[(size_t)b * 32 + tid];
        m[tid] = s * (1.0f / (float)N_TOK);
    }
    __syncthreads();
    if (tid == 0) {
        float L = 0.0f;
        for (int e = 0; e < N_EXP; ++e) { float d = m[e] - (1.0f / (float)N_EXP); L += d * d; }
        out_loss[0] = L;   // mean((m-1/E)^2)*E == sum((m-1/E)^2)
        int s = 0;
        for (int e = 0; e < N_EXP; ++e) { offsets[e] = s; s += counts[e]; }
        s = N_TOK;
        for (int e = 0; e < N_EXP; ++e) { offsets[32 + e] = s; s += counts[32 + e]; }
        for (int i = 0; i < 64; ++i) cursors[i] = offsets[i];
    }
}

// ---------------- bucket scatter ----------------
__global__ __launch_bounds__(256) void k_scatter(const int* __restrict__ topk_idx,
                                                 const float* __restrict__ topk_prob,
                                                 int* __restrict__ cursors,
                                                 int* __restrict__ assign_tok,
                                                 float* __restrict__ assign_prob) {
    const int t = blockIdx.x * 256 + threadIdx.x;
    const int   e0 = topk_idx[2 * t],     e1 = topk_idx[2 * t + 1];
    const float p0 = topk_prob[2 * t],    p1 = topk_prob[2 * t + 1];
    int pos0 = atomicAdd(&cursors[e0], 1);
    assign_tok[pos0] = t;  assign_prob[pos0] = p0;
    int pos1 = atomicAdd(&cursors[32 + e1], 1);
    assign_tok[pos1] = t;  assign_prob[pos1] = p1;
}

// ---------------- routed expert GEMM ----------------
// Wave w of each block owns one fixed (expert, 16-wide N-tile): its B panel
// (16 rows x 256 K = 128 VGPRs) is loaded from L2 ONCE, then the block loops
// over this expert's token tiles (strided by CHUNKS), streaming only the
// 16x256 A panel through LDS per tile. Cuts L2 weight traffic ~8x.
template <int ACCUM>
__global__ __launch_bounds__(512) void k_expert(const float* __restrict__ x,
                                                const float* __restrict__ ew,
                                                const float* __restrict__ eb,
                                                const int* __restrict__ counts,
                                                const int* __restrict__ offsets,
                                                const int* __restrict__ assign_tok,
                                                const float* __restrict__ assign_prob,
                                                float* __restrict__ out) {
    const int slot = ACCUM;
    const int e    = blockIdx.y;
    const int cnt  = counts[slot * 32 + e];
    if ((int)blockIdx.x * 16 >= cnt) return;       // block-uniform: no tiles for this chunk
    const int base = offsets[slot * 32 + e];

    __shared__ float xs[16][260];            // pad 4 -> conflict-free ds_load_b64, 16B-aligned rows
    __shared__ int   tok_s[16];
    __shared__ float prob_s[16];

    const int tid  = threadIdx.x;
    const int w    = tid >> 5;               // wave -> N-tile
    const int l    = tid & 31;
    const int half = l >> 4;
    const int col  = l & 15;
    const int nbase = w * 16;

    // --- load loop-invariant B panel into registers (64 x v2f = 128 VGPRs) ---
    const float* pW = ew + (size_t)(e * D_OUT + nbase + col) * D_IN + 2 * half;
    v2f breg[D_IN / 4];
#pragma unroll
    for (int kk = 0; kk < D_IN / 4; ++kk)
        breg[kk] = *(const v2f*)(pW + kk * 4);

    const float bias = eb[e * D_OUT + nbase + col];
    const float* pA  = &xs[col][2 * half];

    for (int tile = blockIdx.x; tile * 16 < cnt; tile += CHUNKS) {
        const int rows = min(16, cnt - tile * 16);
        __syncthreads();                      // previous tile's reads of xs are done
        if (tid < 16) {
            const int gi = base + tile * 16 + ((tid < rows) ? tid : 0);  // dup row0 for padding
            tok_s[tid]  = assign_tok[gi];
            prob_s[tid] = (tid < rows) ? assign_prob[gi] : 0.0f;
        }
        __syncthreads();
        // cooperative stage of the 16x256 A-panel (gathered token rows)
        for (int i = tid; i < 16 * 64; i += 512) {
            const int row = i >> 6, c4 = i & 63;
            const float4* xr = (const float4*)(x + (size_t)tok_s[row] * D_IN);
            *(float4*)&xs[row][c4 * 4] = xr[c4];
        }
        __syncthreads();

        v8f c = {};
#pragma unroll
        for (int kk = 0; kk < D_IN / 4; ++kk) {
            v2f a = *(const v2f*)(pA + kk * 4);   // ds_load_b64, bank-conflict-free
            c = __builtin_amdgcn_wmma_f32_16x16x4_f32(false, a, false, breg[kk],
                                                      (short)0, c, false, false);
        }

#pragma unroll
        for (int r = 0; r < 8; ++r) {
            const int M = r + 8 * half;
            if (M < rows) {
                const float val = (c[r] + bias) * prob_s[M];
                float* po = out + (size_t)tok_s[M] * D_OUT + nbase + col;
                if (ACCUM) *po += val;        // slot1: unique token per entry, slot0 already done
                else       *po  = val;        // slot0: every token written exactly once
            }
        }
    }
}

extern "C" void kernel_launch(void* const* d_in, const int* in_sizes, int n_in,
                              void* d_out, int out_size, void* d_ws, size_t ws_size,
                              hipStream_t stream) {
    const float* x   = (const float*)d_in[0];
    const float* nz  = (const float*)d_in[1];
    const float* gw  = (const float*)d_in[2];
    const float* gb  = (const float*)d_in[3];
    const float* ew  = (const float*)d_in[4];
    const float* eb  = (const float*)d_in[5];
    float* out = (float*)d_out;

    char* ws = (char*)d_ws;
    int*   topk_idx     = (int*)  (ws + 0);
    float* topk_prob    = (float*)(ws + 256 * 1024);
    int*   assign_tok   = (int*)  (ws + 512 * 1024);
    float* assign_prob  = (float*)(ws + 768 * 1024);
    int*   counts       = (int*)  (ws + 1024 * 1024);
    int*   offsets      = (int*)  (ws + 1024 * 1024 + 256);
    int*   cursors      = (int*)  (ws + 1024 * 1024 + 512);
    float* probsum_part = (float*)(ws + 1024 * 1024 + 768);

    k_init<<<1, 64, 0, stream>>>(counts);
    k_gate<<<N_TOK / 128, 256, 0, stream>>>(x, nz, gw, gb, topk_idx, topk_prob, counts, probsum_part);
    k_scan<<<1, 64, 0, stream>>>(counts, offsets, cursors, probsum_part, out + (size_t)N_TOK * D_OUT);
    k_scatter<<<N_TOK / 256, 256, 0, stream>>>(topk_idx, topk_prob, cursors, assign_tok, assign_prob);
    dim3 gg(CHUNKS, N_EXP);
    k_expert<0><<<gg, 512, 0, stream>>>(x, ew, eb, counts, offsets, assign_tok, assign_prob, out);
    k_expert<1><<<gg, 512, 0, stream>>>(x, ew, eb, counts, offsets, assign_tok, assign_prob, out);
}